// Cross_attention_multi_76115410419869
// MI455X (gfx1250) — compile-verified
//
#include <hip/hip_runtime.h>

// ---------------------------------------------------------------------------
// Cross_attention_multi for MI455X (gfx1250, wave32, WMMA)
// Fuses: att = Ux^T*Uy (per ch), nz=count_nonzero(att), corr = Uxd*(att/nz)^T,
// fold9, LeakyReLU(0.2), out = act*z + z.  att (170MB) never hits HBM.
// ---------------------------------------------------------------------------

typedef __attribute__((ext_vector_type(2))) float v2f;
typedef __attribute__((ext_vector_type(8))) float v8f;

#define S1 186624   // 36*72*72  (xq / z spatial)
#define S2 23328    // 18*36*36  (yk / xd spatial)
#define COUT 64

static __device__ __forceinline__ v8f wmma4(v2f a, v2f b, v8f c) {
  // V_WMMA_F32_16X16X4_F32 : D = A(16x4) x B(4x16) + C(16x16), all fp32
  return __builtin_amdgcn_wmma_f32_16x16x4_f32(
      /*neg_a=*/false, a, /*neg_b=*/false, b,
      /*c_mod=*/(short)0, c, /*reuse_a=*/false, /*reuse_b=*/false);
}

// ---------------- pointwise 1x1x1 conv: out[o][s] = b[o] + sum_c w[o][c]*in[c][s]
template <int CIN>
__global__ void proj_kernel(const float* __restrict__ in,
                            const float* __restrict__ w,
                            const float* __restrict__ b,
                            float* __restrict__ out, int S) {
  __shared__ float ws[COUT * CIN];
  __shared__ float bs[COUT];
  for (int i = threadIdx.x; i < COUT * CIN; i += blockDim.x) ws[i] = w[i];
  if (threadIdx.x < COUT) bs[threadIdx.x] = b[threadIdx.x];
  __syncthreads();
  int s = blockIdx.x * blockDim.x + threadIdx.x;
  if (s >= S) return;
  float xv[CIN];
#pragma unroll
  for (int c = 0; c < CIN; ++c) xv[c] = in[(size_t)c * S + s];
  for (int o = 0; o < COUT; ++o) {
    float acc = bs[o];
#pragma unroll
    for (int c = 0; c < CIN; ++c) acc += ws[o * CIN + c] * xv[c];
    out[(size_t)o * S + s] = acc;
  }
}

// ---------------- AvgPool3d k=3 s=2 p=1, count_include_pad=False on (64,36,72,72)
__global__ void avgpool_kernel(const float* __restrict__ z, float* __restrict__ xd) {
  int idx = blockIdx.x * blockDim.x + threadIdx.x;
  if (idx >= COUT * 18 * 36 * 36) return;
  int ow = idx % 36; int t = idx / 36;
  int oh = t % 36; t /= 36;
  int od = t % 18; int c = t / 18;
  int d0 = max(2 * od - 1, 0), d1 = min(2 * od + 1, 35);
  int h0 = max(2 * oh - 1, 0), h1 = min(2 * oh + 1, 71);
  int w0 = max(2 * ow - 1, 0), w1 = min(2 * ow + 1, 71);
  float sum = 0.f;
  for (int d = d0; d <= d1; ++d)
    for (int h = h0; h <= h1; ++h)
      for (int w = w0; w <= w1; ++w)
        sum += z[((size_t)c * 36 + d) * 5184 + h * 72 + w];
  float cnt = (float)((d1 - d0 + 1) * (h1 - h0 + 1) * (w1 - w0 + 1));
  xd[idx] = sum / cnt;
}

// ---------------- fused attention kernel ----------------
// grid: (mt=0..35, c=0..63), 128 threads (4 waves of 32).
// Block computes patches l in [mt*64, mt*64+64) for channel c.
// m (=288, the Uy/Uxd patch dim) processed in 3 chunks of 96.
__global__ __launch_bounds__(128) void fused_attn_kernel(
    const float* __restrict__ xq, const float* __restrict__ yk,
    const float* __restrict__ xd, const float* __restrict__ z,
    float* __restrict__ out) {
  __shared__ float a1t[64 * 84];    // Ux tile, transposed: [l][k], k padded 81->84 (zeros)
  __shared__ float uyt[96 * 84];    // Uy chunk, transposed: [m][k]
  __shared__ float uxds[96 * 96];   // Uxd chunk: [k][m], k rows 81..95 zero
  __shared__ float atts[64 * 96];   // att tile: [l][m]
  __shared__ float nzp[128];
  __shared__ float nzscale[64];

  const int mt = blockIdx.x;
  const int c = blockIdx.y;
  const int tid = threadIdx.x;
  const int lane = tid & 31;
  const int wv = tid >> 5;              // wave 0..3
  const int lrow = lane & 15;
  const int khalf = (lane >> 4) << 1;   // 0 or 2 : K sub-pair per half-wave
  const int rhalf = (lane >> 4) << 3;   // 0 or 8 : M offset of C rows per half-wave

  // ---- load Ux tile (once): a1t[l][k] = Ux(c, k, mt*64+l)
  for (int idx = tid; idx < 64 * 84; idx += 128) {
    int l = idx / 84, k = idx % 84;
    float v = 0.f;
    if (k < 81) {
      int lg = mt * 64 + l;
      int ph = lg / 576, pw = lg % 576;
      int i = k / 9, j = k % 9;
      v = xq[(size_t)c * S1 + (size_t)(ph * 9 + i) * 5184 + pw * 9 + j];
    }
    a1t[idx] = v;
  }

  const v8f vzero = {0.f, 0.f, 0.f, 0.f, 0.f, 0.f, 0.f, 0.f};
  v8f acc2[6];
#pragma unroll
  for (int t = 0; t < 6; ++t) acc2[t] = vzero;

  float nzcnt = 0.f;
  const int nrow = tid >> 1;   // 0..63
  const int nhalf = tid & 1;

  for (int ch = 0; ch < 3; ++ch) {
    __syncthreads();  // prior chunk's consumers of uyt/uxds/atts are done
    // ---- fill Uy chunk (transposed) and Uxd chunk
    for (int idx = tid; idx < 96 * 84; idx += 128) {
      int m = idx / 84, k = idx % 84;
      float v = 0.f;
      if (k < 81) {
        int mg = ch * 96 + m;
        int ph2 = mg / 144, pw2 = mg % 144;
        int i = k / 9, j = k % 9;
        v = yk[(size_t)c * S2 + (size_t)(ph2 * 9 + i) * 1296 + pw2 * 9 + j];
      }
      uyt[idx] = v;
    }
    for (int idx = tid; idx < 96 * 96; idx += 128) {
      int k = idx / 96, m = idx % 96;
      float v = 0.f;
      if (k < 81) {
        int mg = ch * 96 + m;
        int ph2 = mg / 144, pw2 = mg % 144;
        int i = k / 9, j = k % 9;
        v = xd[(size_t)c * S2 + (size_t)(ph2 * 9 + i) * 1296 + pw2 * 9 + j];
      }
      uxds[idx] = v;
    }
    __syncthreads();

    // ---- GEMM1: att(l,m) = sum_k Ux(k,l)*Uy(k,m). Wave wv: rows 16wv..16wv+15.
    const float* ap0 = &a1t[(wv * 16 + lrow) * 84 + khalf];
    for (int nt = 0; nt < 6; ++nt) {
      v8f acc = vzero;
      const float* bp0 = &uyt[(nt * 16 + lrow) * 84 + khalf];
      for (int kk = 0; kk < 21; ++kk) {
        v2f a = *(const v2f*)(ap0 + 4 * kk);
        v2f b = *(const v2f*)(bp0 + 4 * kk);
        acc = wmma4(a, b, acc);
      }
      int colb = nt * 16 + lrow;
      int rowb = wv * 16 + rhalf;
#pragma unroll
      for (int r = 0; r < 8; ++r) atts[(rowb + r) * 96 + colb] = acc[r];
    }
    __syncthreads();

    // ---- nonzero-count partial (2 threads per att row, register-accumulated)
    {
      const float* rp = &atts[nrow * 96 + nhalf * 48];
      for (int m = 0; m < 48; ++m) nzcnt += (rp[m] != 0.f) ? 1.f : 0.f;
    }

    // ---- GEMM2 accumulate: corr(k,l) += sum_m Uxd(k,m)*att(l,m).
    // Wave wv owns l-tile wv (N cols), all 6 k-tiles (M rows 0..95).
    for (int mm = 0; mm < 24; ++mm) {
      v2f b = *(const v2f*)(&atts[(wv * 16 + lrow) * 96 + 4 * mm + khalf]);
#pragma unroll
      for (int kt = 0; kt < 6; ++kt) {
        v2f a = *(const v2f*)(&uxds[(kt * 16 + lrow) * 96 + 4 * mm + khalf]);
        acc2[kt] = wmma4(a, b, acc2[kt]);
      }
    }
  }

  __syncthreads();
  nzp[nhalf * 64 + nrow] = nzcnt;
  __syncthreads();
  if (tid < 64) nzscale[tid] = 1.f / (nzp[tid] + nzp[64 + tid] + 1e-5f);
  __syncthreads();

  // ---- epilogue: scale by 1/nz, fold9, LeakyReLU(0.2), residual with z
  const int lloc = wv * 16 + lrow;
  const float sc = nzscale[lloc];
  const int lg = mt * 64 + lloc;
  const int ph = lg / 576, pw = lg % 576;
#pragma unroll
  for (int kt = 0; kt < 6; ++kt) {
#pragma unroll
    for (int r = 0; r < 8; ++r) {
      int k = kt * 16 + rhalf + r;
      if (k < 81) {
        float corr = acc2[kt][r] * sc;
        float act = corr > 0.f ? corr : 0.2f * corr;
        int i = k / 9, j = k % 9;
        size_t idx = (size_t)c * S1 + (size_t)(ph * 9 + i) * 5184 + pw * 9 + j;
        float zv = z[idx];
        out[idx] = act * zv + zv;
      }
    }
  }
}

// ---------------------------------------------------------------------------
extern "C" void kernel_launch(void* const* d_in, const int* in_sizes, int n_in,
                              void* d_out, int out_size, void* d_ws, size_t ws_size,
                              hipStream_t stream) {
  const float* x = (const float*)d_in[0];      // (1,32,36,72,72)
  const float* y = (const float*)d_in[1];      // (1,64,18,36,36)
  const float* z = (const float*)d_in[2];      // (1,64,36,72,72)
  const float* w_img = (const float*)d_in[3];  // (64,32)
  const float* b_img = (const float*)d_in[4];  // (64,)
  const float* w_fea = (const float*)d_in[5];  // (64,64)
  const float* b_fea = (const float*)d_in[6];  // (64,)
  float* out = (float*)d_out;

  // workspace: xq (64*S1) + yk (64*S2) + xd (64*S2)  = 59,719,680 bytes
  float* xq = (float*)d_ws;
  float* yk = xq + (size_t)COUT * S1;
  float* xdp = yk + (size_t)COUT * S2;

  proj_kernel<32><<<dim3(S1 / 256), dim3(256), 0, stream>>>(x, w_img, b_img, xq, S1);
  proj_kernel<64><<<dim3((S2 + 255) / 256), dim3(256), 0, stream>>>(y, w_fea, b_fea, yk, S2);
  avgpool_kernel<<<dim3((COUT * S2 + 255) / 256), dim3(256), 0, stream>>>(z, xdp);
  fused_attn_kernel<<<dim3(36, 64), dim3(128), 0, stream>>>(xq, yk, xdp, z, out);
}